// Classifier_39118562132299
// MI455X (gfx1250) — compile-verified
//
#include <hip/hip_runtime.h>

#define N_NODES    100000
#define N_EDGES    1600000
#define HIDDEN     128
#define N_CLASSES  10
#define NUM_GRAPHS 128

typedef __attribute__((ext_vector_type(2))) float v2f;
typedef __attribute__((ext_vector_type(8))) float v8f;

// ---------------- zero accumulators ----------------
__global__ void k_zero(float* __restrict__ p, int n4) {
    int i = blockIdx.x * blockDim.x + threadIdx.x;
    if (i < n4) ((float4*)p)[i] = make_float4(0.f, 0.f, 0.f, 0.f);
}

// ---------------- in_deg = segment_sum(1, dst) ----------------
__global__ void k_indeg(const int* __restrict__ dst, float* __restrict__ in_deg) {
    int e = blockIdx.x * blockDim.x + threadIdx.x;
    if (e < N_EDGES) atomicAdd(&in_deg[dst[e]], 1.0f);
}

// ---------------- s1 = segment_sum(in_deg[src], dst) ----------------
__global__ void k_s1(const int* __restrict__ src, const int* __restrict__ dst,
                     const float* __restrict__ in_deg, float* __restrict__ s1) {
    int e = blockIdx.x * blockDim.x + threadIdx.x;
    if (e < N_EDGES) atomicAdd(&s1[dst[e]], in_deg[src[e]]);
}

// ---------------- agg1[n] = has_in ? s1/deg : in_deg ; counts ----------------
__global__ void k_agg1(const float* __restrict__ in_deg, const float* __restrict__ s1,
                       const int* __restrict__ gid,
                       float* __restrict__ agg1, float* __restrict__ counts) {
    int n = blockIdx.x * blockDim.x + threadIdx.x;
    if (n < N_NODES) {
        float d = in_deg[n];
        agg1[n] = (d > 0.f) ? s1[n] / d : d;   // else-branch: h = in_deg (== d)
        atomicAdd(&counts[gid[n]], 1.0f);
    }
}

// ---- s2 = segment_sum(h1[src], dst); h1 reconstructed on the fly (rank-1 layer 1) ----
// one wave32 per edge, each lane owns 4 features; skip relu-zeros (atomic of 0 is a no-op)
__global__ void k_edge_scatter(const int* __restrict__ src, const int* __restrict__ dst,
                               const float* __restrict__ agg1,
                               const float* __restrict__ W1, const float* __restrict__ b1,
                               float* __restrict__ s2) {
    int t = blockIdx.x * blockDim.x + threadIdx.x;
    int e = t >> 5;
    if (e >= N_EDGES) return;
    int lane = t & 31;
    int f    = lane * 4;
    int s = src[e];
    int d = dst[e];
    float a = agg1[s];                    // broadcast load (same addr across wave)
    float4 w  = *(const float4*)(W1 + f);
    float4 bb = *(const float4*)(b1 + f);
    float4 h;
    h.x = fmaxf(fmaf(a, w.x, bb.x), 0.f);
    h.y = fmaxf(fmaf(a, w.y, bb.y), 0.f);
    h.z = fmaxf(fmaf(a, w.z, bb.z), 0.f);
    h.w = fmaxf(fmaf(a, w.w, bb.w), 0.f);
    float* out = s2 + (size_t)d * HIDDEN + f;
    if (h.x != 0.f) atomicAdd(out + 0, h.x);
    if (h.y != 0.f) atomicAdd(out + 1, h.y);
    if (h.z != 0.f) atomicAdd(out + 2, h.z);
    if (h.w != 0.f) atomicAdd(out + 3, h.w);
}

// ---- layer 2 GEMM via V_WMMA_F32_16X16X4_F32, fused relu + graph pooling ----
// one block per 16-node tile; 8 waves -> 8 column tiles of 16; K=128 in steps of 4
__global__ void __launch_bounds__(256)
k_gcn2_wmma(const float* __restrict__ in_deg, const float* __restrict__ agg1,
            const float* __restrict__ s2,
            const float* __restrict__ W1, const float* __restrict__ b1,
            const float* __restrict__ W2, const float* __restrict__ b2,
            const int* __restrict__ gid, float* __restrict__ hg) {
    __shared__ float As[16][HIDDEN + 4];   // +4 pad -> conflict-free column reads
    const int tile = blockIdx.x;
    const int t    = threadIdx.x;

    // stage h_agg (16 x 128) into LDS: has_in ? s2/deg : relu(agg1*W1+b1)
    {
        int nl   = t >> 4;          // 0..15 node-in-tile
        int f0   = (t & 15) * 8;    // 8 features per thread
        int node = tile * 16 + nl;
        float d    = in_deg[node];
        float a1   = agg1[node];
        bool  hin  = d > 0.f;
        float inv  = hin ? 1.f / d : 0.f;
        const float* row = s2 + (size_t)node * HIDDEN;
        #pragma unroll
        for (int i = 0; i < 8; ++i) {
            int f = f0 + i;
            As[nl][f] = hin ? row[f] * inv
                            : fmaxf(fmaf(a1, W1[f], b1[f]), 0.f);
        }
    }
    __syncthreads();

    const int wave = t >> 5;        // 0..7 -> N-tile
    const int lane = t & 31;
    const int n0   = wave * 16;
    const int lo   = lane & 15;     // A: M ; B/D: N
    const int hi   = lane >> 4;     // K-quad half select

    v8f c = {};
    #pragma unroll 4
    for (int k0 = 0; k0 < HIDDEN; k0 += 4) {
        int kb = k0 + (hi << 1);    // lanes 0-15: K=k0,k0+1 ; lanes 16-31: K=k0+2,k0+3
        v2f a, b;
        a.x = As[lo][kb];
        a.y = As[lo][kb + 1];
        int n = n0 + lo;
        b.x = W2[kb * HIDDEN + n];
        b.y = W2[(kb + 1) * HIDDEN + n];
        c = __builtin_amdgcn_wmma_f32_16x16x4_f32(
                /*neg_a=*/false, a, /*neg_b=*/false, b,
                /*c_mod=*/(short)0, c, /*reuse_a=*/false, /*reuse_b=*/false);
    }

    // epilogue: +b2, relu, pool into hg[graph, feature]
    // D layout: VGPR r -> M = hi*8 + r, N = lo. The 8 nodes this lane pools for
    // are consecutive -> hoist their graph ids as two int4 loads (outside the
    // divergent zero-skip branches, so addresses are precomputed once).
    float bv = b2[n0 + lo];
    const int* grow = gid + tile * 16 + (hi << 3);
    int4 g0 = *(const int4*)(grow);
    int4 g1 = *(const int4*)(grow + 4);
    int gg[8] = {g0.x, g0.y, g0.z, g0.w, g1.x, g1.y, g1.z, g1.w};
    float* hgcol = hg + n0 + lo;
    #pragma unroll
    for (int r = 0; r < 8; ++r) {
        float v = fmaxf(c[r] + bv, 0.f);
        if (v != 0.f)
            atomicAdd(hgcol + (size_t)gg[r] * HIDDEN, v);
    }
}

// ---------------- out = (hg / max(counts,1)) @ Wc + bc ----------------
__global__ void k_final(const float* __restrict__ hg, const float* __restrict__ counts,
                        const float* __restrict__ Wc, const float* __restrict__ bc,
                        float* __restrict__ out) {
    int t = blockIdx.x * blockDim.x + threadIdx.x;
    if (t >= NUM_GRAPHS * N_CLASSES) return;
    int g  = t / N_CLASSES;
    int cc = t % N_CLASSES;
    float inv = 1.0f / fmaxf(counts[g], 1.0f);
    float acc = 0.f;
    #pragma unroll 8
    for (int j = 0; j < HIDDEN; ++j)
        acc += hg[g * HIDDEN + j] * Wc[j * N_CLASSES + cc];
    out[t] = acc * inv + bc[cc];
}

extern "C" void kernel_launch(void* const* d_in, const int* in_sizes, int n_in,
                              void* d_out, int out_size, void* d_ws, size_t ws_size,
                              hipStream_t stream) {
    const int*   src = (const int*)  d_in[0];
    const int*   dst = (const int*)  d_in[1];
    const int*   gid = (const int*)  d_in[2];
    const float* W1  = (const float*)d_in[3];
    const float* b1  = (const float*)d_in[4];
    const float* W2  = (const float*)d_in[5];
    const float* b2  = (const float*)d_in[6];
    const float* Wc  = (const float*)d_in[7];
    const float* bc  = (const float*)d_in[8];
    float* out = (float*)d_out;

    // workspace layout: [zeroed accumulators ...][agg1]
    float* ws     = (float*)d_ws;
    float* in_deg = ws;                                   // N
    float* s1     = in_deg + N_NODES;                     // N
    float* counts = s1 + N_NODES;                         // 128
    float* hg     = counts + NUM_GRAPHS;                  // 128*128
    float* s2     = hg + NUM_GRAPHS * HIDDEN;             // N*128
    float* agg1   = s2 + (size_t)N_NODES * HIDDEN;        // N (fully overwritten)

    size_t zf = 2ull * N_NODES + NUM_GRAPHS + (size_t)NUM_GRAPHS * HIDDEN
              + (size_t)N_NODES * HIDDEN;                 // 13,016,512 floats
    int n4 = (int)(zf / 4);

    k_zero        <<<(n4 + 255) / 256,                 256, 0, stream>>>(ws, n4);
    k_indeg       <<<(N_EDGES + 255) / 256,            256, 0, stream>>>(dst, in_deg);
    k_s1          <<<(N_EDGES + 255) / 256,            256, 0, stream>>>(src, dst, in_deg, s1);
    k_agg1        <<<(N_NODES + 255) / 256,            256, 0, stream>>>(in_deg, s1, gid, agg1, counts);
    k_edge_scatter<<<(int)(((long long)N_EDGES * 32 + 255) / 256), 256, 0, stream>>>(src, dst, agg1, W1, b1, s2);
    k_gcn2_wmma   <<<N_NODES / 16,                     256, 0, stream>>>(in_deg, agg1, s2, W1, b1, W2, b2, gid, hg);
    k_final       <<<(NUM_GRAPHS * N_CLASSES + 255) / 256, 256, 0, stream>>>(hg, counts, Wc, bc, out);
}